// TvaeElboLoss_12111807774852
// MI455X (gfx1250) — compile-verified
//
#include <hip/hip_runtime.h>
#include <stdint.h>

#define B_ROWS     65536
#define N_CONT     20
#define N_CLUST    10
#define N_CAT      20
#define N_CATS     16
#define CONT_BLK   11
#define CONT_DIM   220
#define D_DIM      540
#define LATENT     128

#define TPB        256
#define MAXGRID    2048

typedef __attribute__((ext_vector_type(2))) float v2f;
typedef __attribute__((ext_vector_type(8))) float v8f;

// ---------------------------------------------------------------------------
// Exact f32 wave32 reduction assisted by the CDNA5 matrix pipe.
//   A = ones(16x4)   (both A VGPRs = 1.0 in every lane)
//   B = lane partial in VGPR0, 0 in VGPR1  -> the 32 partials occupy 32
//       distinct (k,n) slots of the 4x16 B matrix, the rest are zero.
//   D[m,n] = sum_k A[m,k]*B[k,n] = colsum_n  (independent of m), f32 RNE FMA.
// D's VGPR0 holds row M=0 in lanes 0-15 and row M=8 in lanes 16-31 — both are
// the same column sums, so summing d[0] over all 32 lanes counts every lane
// partial exactly twice. Halve at the end. Layout-robust: only relies on the
// documented C/D layout.
// ---------------------------------------------------------------------------
__device__ __forceinline__ float wave_sum32(float p) {
    v2f a; a[0] = 1.0f; a[1] = 1.0f;
    v2f b; b[0] = p;    b[1] = 0.0f;
    v8f c = {};
    v8f d = __builtin_amdgcn_wmma_f32_16x16x4_f32(
        /*neg_a=*/false, a, /*neg_b=*/false, b,
        /*c_mod=*/(short)0, c, /*reuse_a=*/false, /*reuse_b=*/false);
    float s = d[0];
    s += __shfl_xor(s, 16, 32);
    s += __shfl_xor(s,  8, 32);
    s += __shfl_xor(s,  4, 32);
    s += __shfl_xor(s,  2, 32);
    s += __shfl_xor(s,  1, 32);
    return 0.5f * s;
}

// Sparse categorical CE term from register arrays: lse(q) - q[argmax(r)].
// Strict '>' keeps the first max, matching jnp.argmax tie-breaking.
template <int K>
__device__ __forceinline__ float ce_term(const float* r, const float* q) {
    float mr = r[0], pick = q[0];
#pragma unroll
    for (int k = 1; k < K; ++k) {
        bool t = r[k] > mr;
        mr   = t ? r[k] : mr;
        pick = t ? q[k] : pick;
    }
    float mg = q[0];
#pragma unroll
    for (int k = 1; k < K; ++k) mg = fmaxf(mg, q[k]);
    float se = 0.0f;
#pragma unroll
    for (int k = 0; k < K; ++k) se += __expf(q[k] - mg);
    return mg + __logf(se) - pick;
}

__global__ __launch_bounds__(TPB)
void elbo_partial(const float* __restrict__ xr, const float* __restrict__ xg,
                  const float* __restrict__ sig,
                  const float* __restrict__ mean, const float* __restrict__ lvar,
                  float* __restrict__ partials)
{
    // LDS staging: 4 aligned 16B chunks per thread per matrix, laid out
    // chunk-major so each async b128 write and each ds_load_b128 readback has
    // a 16B lane stride (every LDS bank touched exactly twice per wave).
    __shared__ float4 s_r[4 * TPB];
    __shared__ float4 s_g[4 * TPB];
    __shared__ float red1[TPB / 32], red2[TPB / 32];

    const unsigned tid = threadIdx.x;
    const unsigned gsz = gridDim.x * TPB;
    float s1 = 0.0f;   // cont_recon residual + ce_cont + ce_cat
    float s2 = 0.0f;   // sum(1 + lv - mean^2 - exp(lv));  kld = -0.5 * s2

    // ---- categorical CE: stage 16-float groups via CDNA5 async DMA --------
    const unsigned NGCAT = (unsigned)B_ROWS * N_CAT;           // 1,310,720
    const unsigned iters = (NGCAT + gsz - 1) / gsz;
    unsigned g = blockIdx.x * TPB + tid;
    for (unsigned it = 0; it < iters; ++it, g += gsz) {
        const bool act = g < NGCAT;
        if (act) {
            const unsigned brow = g / N_CAT, f = g % N_CAT;
            const size_t base = (size_t)brow * D_DIM + CONT_DIM + (size_t)f * N_CATS;
            const float* pr = xr + base;   // 16B aligned (base*4 % 16 == 0)
            const float* pg = xg + base;
#pragma unroll
            for (int i = 0; i < 4; ++i) {
                // low 32 bits of a generic LDS pointer == workgroup LDS offset
                unsigned lr = (unsigned)(size_t)&s_r[i * TPB + tid];
                unsigned lg = (unsigned)(size_t)&s_g[i * TPB + tid];
                unsigned long long gr = (unsigned long long)(pr + 4 * i);
                unsigned long long gq = (unsigned long long)(pg + 4 * i);
                asm volatile("global_load_async_to_lds_b128 %0, %1, off"
                             :: "v"(lr), "v"(gr) : "memory");
                asm volatile("global_load_async_to_lds_b128 %0, %1, off"
                             :: "v"(lg), "v"(gq) : "memory");
            }
        }
        // Per-wave wait: each thread reads only the LDS chunks it requested.
        asm volatile("s_wait_asynccnt 0" ::: "memory");
        if (act) {
            float r[N_CATS], q[N_CATS];
#pragma unroll
            for (int i = 0; i < 4; ++i) {
                float4 t = s_r[i * TPB + tid];
                r[4*i+0]=t.x; r[4*i+1]=t.y; r[4*i+2]=t.z; r[4*i+3]=t.w;
                t = s_g[i * TPB + tid];
                q[4*i+0]=t.x; q[4*i+1]=t.y; q[4*i+2]=t.z; q[4*i+3]=t.w;
            }
            s1 += ce_term<N_CATS>(r, q);
        }
    }

    // ---- continuous blocks: alpha residual + 10-way cluster CE ------------
    const unsigned NGC = (unsigned)B_ROWS * N_CONT;
    for (unsigned h = blockIdx.x * TPB + tid; h < NGC; h += gsz) {
        const unsigned brow = h / N_CONT, f = h % N_CONT;
        const size_t base = (size_t)brow * D_DIM + (size_t)f * CONT_BLK;
        const float* pr = xr + base;
        const float* pg = xg + base;
        const float sd = sig[f * CONT_BLK];
        const float e  = pr[0] - tanhf(pg[0]);
        s1 += e * e / (2.0f * sd * sd);
        float r[N_CLUST], q[N_CLUST];
#pragma unroll
        for (int k = 0; k < N_CLUST; ++k) { r[k] = pr[1 + k]; q[k] = pg[1 + k]; }
        s1 += ce_term<N_CLUST>(r, q);
    }

    // ---- KL divergence: coalesced float4 stream ---------------------------
    const unsigned NL4 = (unsigned)(B_ROWS * LATENT / 4);      // 2,097,152
    const float4* m4 = (const float4*)mean;
    const float4* v4 = (const float4*)lvar;
    for (unsigned i = blockIdx.x * TPB + tid; i < NL4; i += gsz) {
        const float4 m = m4[i], l = v4[i];
        s2 += (1.0f + l.x - m.x * m.x - __expf(l.x));
        s2 += (1.0f + l.y - m.y * m.y - __expf(l.y));
        s2 += (1.0f + l.z - m.z * m.z - __expf(l.z));
        s2 += (1.0f + l.w - m.w * m.w - __expf(l.w));
    }

    // ---- block reduction (all lanes converged -> EXEC all-1s for WMMA) ----
    const float w1 = wave_sum32(s1);
    const float w2 = wave_sum32(s2);
    const unsigned lane = tid & 31, wid = tid >> 5;
    if (lane == 0) { red1[wid] = w1; red2[wid] = w2; }
    __syncthreads();
    if (tid == 0) {
        float a = 0.0f, b = 0.0f;
#pragma unroll
        for (int i = 0; i < TPB / 32; ++i) { a += red1[i]; b += red2[i]; }
        partials[2 * blockIdx.x]     = a;
        partials[2 * blockIdx.x + 1] = b;
    }
}

__global__ __launch_bounds__(TPB)
void elbo_final(const float* __restrict__ partials, int nblk,
                const float* __restrict__ sig,
                const float* __restrict__ loss_factor,
                float* __restrict__ out)
{
    __shared__ float red1[TPB / 32], red2[TPB / 32];
    float s1 = 0.0f, s2 = 0.0f;
    for (int i = threadIdx.x; i < nblk; i += TPB) {
        s1 += partials[2 * i];
        s2 += partials[2 * i + 1];
    }
    const float w1 = wave_sum32(s1);
    const float w2 = wave_sum32(s2);
    const unsigned lane = threadIdx.x & 31, wid = threadIdx.x >> 5;
    if (lane == 0) { red1[wid] = w1; red2[wid] = w2; }
    __syncthreads();
    if (threadIdx.x == 0) {
        float a = 0.0f, b = 0.0f;
#pragma unroll
        for (int i = 0; i < TPB / 32; ++i) { a += red1[i]; b += red2[i]; }
        float slog = 0.0f;
        for (int f = 0; f < N_CONT; ++f) slog += __logf(sig[f * CONT_BLK]);
        const float S1  = a + (float)B_ROWS * slog;   // + Bf * sum(log std)
        const float kld = -0.5f * b;
        out[0] = (S1 * loss_factor[0] + kld) / (float)D_DIM;
    }
}

extern "C" void kernel_launch(void* const* d_in, const int* in_sizes, int n_in,
                              void* d_out, int out_size, void* d_ws, size_t ws_size,
                              hipStream_t stream) {
    (void)in_sizes; (void)n_in; (void)out_size;
    const float* xr   = (const float*)d_in[0];
    const float* xg   = (const float*)d_in[1];
    const float* sig  = (const float*)d_in[2];
    const float* mean = (const float*)d_in[3];
    const float* lvar = (const float*)d_in[4];
    const float* lf   = (const float*)d_in[5];

    int grid = MAXGRID;
    const size_t need = (size_t)grid * 2 * sizeof(float);
    if (ws_size < need) {
        grid = (int)(ws_size / (2 * sizeof(float)));
        if (grid < 1) grid = 1;
    }
    float* partials = (float*)d_ws;

    elbo_partial<<<grid, TPB, 0, stream>>>(xr, xg, sig, mean, lvar, partials);
    elbo_final<<<1, TPB, 0, stream>>>(partials, grid, sig, lf, (float*)d_out);
}